// FusedTensorProductScatter_67765993996329
// MI455X (gfx1250) — compile-verified
//
#include <hip/hip_runtime.h>

typedef __attribute__((ext_vector_type(2))) float v2f;
typedef __attribute__((ext_vector_type(8))) float v8f;

#define IN_DIM  40
#define OUT_DIM 40
#define WNUM    576

__global__ void zero_f32_kernel(float* __restrict__ p, int n) {
    int i = blockIdx.x * blockDim.x + threadIdx.x;
    if (i < n) p[i] = 0.0f;
}

// One wave32 per edge (grid-stride). Per edge:
//   d0 (16x16 f32 acc): m0[n] replicated over rows
//   d1 (16x16 f32 acc): rows 0..2 = kdim, cols 0..7 = w  -> m1[w,kdim]
__global__ __launch_bounds__(256) void tp_scatter_wmma_kernel(
    const float* __restrict__ x,      // (N,40)
    const float* __restrict__ eattr,  // (E,4)  s, v0,v1,v2
    const float* __restrict__ ew,     // (E,576)
    const int*   __restrict__ esrc,
    const int*   __restrict__ edst,
    float*       __restrict__ out,    // (N,40), pre-zeroed
    int E)
{
    constexpr float C_A = 0.20412414523193154f;  // 1/sqrt(24)
    constexpr float C_B = 0.11785113019775793f;  // 1/sqrt(72)

    const int lane   = threadIdx.x & 31;
    const int m      = lane & 15;              // A row / B-D column index
    const int hi2    = ((lane >> 4) & 1) << 1; // 0 (lanes 0-15) or 2 (lanes 16-31)
    const int wave   = blockIdx.x * (blockDim.x >> 5) + (threadIdx.x >> 5);
    const int nwaves = gridDim.x * (blockDim.x >> 5);

    for (int e = wave; e < E; e += nwaves) {
        const float* __restrict__ w = ew + (size_t)e * WNUM;

        // Prefetch next edge's 2304B weight block (32 lanes x 72B covers it).
        if (e + nwaves < E) {
            __builtin_prefetch((const char*)(ew + (size_t)(e + nwaves) * WNUM) + lane * 72, 0, 0);
        }

        const int   src = esrc[e];
        const float* __restrict__ xe = x + src * IN_DIM;  // xs0=xe[0..15], xs1[u][k]=xe[16+3u+k]
        const float s  = eattr[e * 4 + 0];
        const float v0 = eattr[e * 4 + 1];
        const float v1 = eattr[e * 4 + 2];
        const float v2 = eattr[e * 4 + 3];
        const float cas  = C_A * s;
        const float vm   = (m == 0) ? v0 : ((m == 1) ? v1 : v2);
        const float cavm = (m < 3) ? (C_A * vm) : 0.0f;   // rank-1 A row scale for W011 term
        const int   mk   = (m < 3) ? m : 0;               // clamped kdim for xs1^T rows

        v8f d0 = {};
        v8f d1 = {};

        // ---- m0 += (C_A*s*xs0) @ W000, K=16 (4 chained WMMAs). B = W000 natural layout.
#pragma unroll
        for (int kc = 0; kc < 4; ++kc) {
            const int k0 = kc * 4 + hi2;
            v2f a, b;
            a[0] = cas * xe[k0];
            a[1] = cas * xe[k0 + 1];
            b[0] = w[k0 * 16 + m];
            b[1] = w[(k0 + 1) * 16 + m];
            d0 = __builtin_amdgcn_wmma_f32_16x16x4_f32(false, a, false, b, (short)0, d0, false, false);
        }

        // ---- m0 += (C_B*dot11) @ W110, K=8 (2 WMMAs). dot11[u] = xs1[u,:].v
#pragma unroll
        for (int kc = 0; kc < 2; ++kc) {
            const int k0 = kc * 4 + hi2;
            const float* r = xe + 16 + k0 * 3;
            v2f a, b;
            a[0] = C_B * (r[0] * v0 + r[1] * v1 + r[2] * v2);
            a[1] = C_B * (r[3] * v0 + r[4] * v1 + r[5] * v2);
            b[0] = w[448 + k0 * 16 + m];
            b[1] = w[448 + (k0 + 1) * 16 + m];
            d0 = __builtin_amdgcn_wmma_f32_16x16x4_f32(false, a, false, b, (short)0, d0, false, false);
        }

        // ---- m1[kdim,w] += C_A*v[kdim] * (xs0 @ W011), K=16. Rank-1 A, B cols 8..15 zeroed.
#pragma unroll
        for (int kc = 0; kc < 4; ++kc) {
            const int k0 = kc * 4 + hi2;
            v2f a, b;
            a[0] = cavm * xe[k0];
            a[1] = cavm * xe[k0 + 1];
            const float b0 = w[256 + k0 * 8 + (m & 7)];
            const float b1 = w[256 + (k0 + 1) * 8 + (m & 7)];
            b[0] = (m < 8) ? b0 : 0.0f;
            b[1] = (m < 8) ? b1 : 0.0f;
            d1 = __builtin_amdgcn_wmma_f32_16x16x4_f32(false, a, false, b, (short)0, d1, false, false);
        }

        // ---- m1[kdim,w] += C_A*s * (xs1^T @ W101), K=8. A rows 3..15 zeroed.
#pragma unroll
        for (int kc = 0; kc < 2; ++kc) {
            const int k0 = kc * 4 + hi2;
            v2f a, b;
            const float a0 = cas * xe[16 + k0 * 3 + mk];
            const float a1 = cas * xe[16 + (k0 + 1) * 3 + mk];
            a[0] = (m < 3) ? a0 : 0.0f;
            a[1] = (m < 3) ? a1 : 0.0f;
            const float b0 = w[384 + k0 * 8 + (m & 7)];
            const float b1 = w[384 + (k0 + 1) * 8 + (m & 7)];
            b[0] = (m < 8) ? b0 : 0.0f;
            b[1] = (m < 8) ? b1 : 0.0f;
            d1 = __builtin_amdgcn_wmma_f32_16x16x4_f32(false, a, false, b, (short)0, d1, false, false);
        }

        // ---- scatter: m0[n] from d0[0] (rows replicated); m1[w,kdim] = d1[kdim] at col w
        const int dst = edst[e];
        float* ob = out + dst * OUT_DIM;
        if (lane < 16) {
            atomicAdd(ob + m, d0[0]);
        }
        if (lane < 8) {
            atomicAdd(ob + 16 + m * 3 + 0, d1[0]);
            atomicAdd(ob + 16 + m * 3 + 1, d1[1]);
            atomicAdd(ob + 16 + m * 3 + 2, d1[2]);
        }
    }
}

extern "C" void kernel_launch(void* const* d_in, const int* in_sizes, int n_in,
                              void* d_out, int out_size, void* d_ws, size_t ws_size,
                              hipStream_t stream) {
    const float* x     = (const float*)d_in[0];
    const float* eattr = (const float*)d_in[1];
    const float* ew    = (const float*)d_in[2];
    const int*   esrc  = (const int*)d_in[3];
    const int*   edst  = (const int*)d_in[4];
    float* out = (float*)d_out;

    const int E = in_sizes[2] / WNUM;   // 320000

    // Output is accumulated with atomics; harness poisons d_out, so zero it first.
    zero_f32_kernel<<<(out_size + 255) / 256, 256, 0, stream>>>(out, out_size);

    // 4096 blocks x 8 waves = 32768 waves, ~10 edges each (grid-stride).
    tp_scatter_wmma_kernel<<<4096, 256, 0, stream>>>(x, eattr, ew, esrc, edst, out, E);
}